// GraphLayerAtt_19868518711636
// MI455X (gfx1250) — compile-verified
//
#include <hip/hip_runtime.h>

typedef __attribute__((ext_vector_type(2))) float v2f;
typedef __attribute__((ext_vector_type(8))) float v8f;

#define B_DIM 8
#define S_DIM 512
#define D_DIM 512
#define E 32
#define LPAD 33   // LDS row stride (floats), odd -> conflict-free across 64 banks

// Load a 16(row) x 32(K) f32 operand (row-major, stride rstride floats) into the
// WMMA 16x16x4-f32 A/B register layout, as 8 K-chunks of v2f.
// Lane l (l<16):  rows m=l,    holds K = {4j+0, 4j+1}
// Lane l (l>=16): rows m=l-16, holds K = {4j+2, 4j+3}
__device__ inline void load_op16x32(const float* __restrict__ base, int rstride, v2f op[8]) {
  const int l = threadIdx.x & 31;
  const int m = l & 15;
  const int h = l >> 4;
  const float* p = base + (size_t)m * rstride + 2 * h;
#pragma unroll
  for (int j = 0; j < 8; ++j)
    op[j] = *(const v2f*)(p + 4 * j);
}

// K=32 accumulation chain: 8x V_WMMA_F32_16X16X4_F32
__device__ inline v8f wmma_chain(const v2f a[8], const v2f b[8], v8f c) {
#pragma unroll
  for (int j = 0; j < 8; ++j)
    c = __builtin_amdgcn_wmma_f32_16x16x4_f32(false, a[j], false, b[j], (short)0, c,
                                              false, false);
  return c;
}

// ---------------------------------------------------------------------------
// Node precompute: outA = X @ Wa^T + ba ; outB = X @ Wb^T
// blockIdx.y == 0 -> x_s path, == 1 -> x_d path. 8 waves/WG, 16 rows/wave.
// ---------------------------------------------------------------------------
__global__ void node_kernel(const float* __restrict__ xs, const float* __restrict__ xd,
                            const float* __restrict__ Wss, const float* __restrict__ bss,
                            const float* __restrict__ Wes,
                            const float* __restrict__ Wdd, const float* __restrict__ bdd,
                            const float* __restrict__ Wed,
                            float* __restrict__ hsb, float* __restrict__ xses,
                            float* __restrict__ hdb, float* __restrict__ xded) {
  const float *X, *Wa, *ba, *Wb;
  float *oA, *oB;
  if (blockIdx.y == 0) { X = xs; Wa = Wss; ba = bss; Wb = Wes; oA = hsb; oB = xses; }
  else                 { X = xd; Wa = Wdd; ba = bdd; Wb = Wed; oA = hdb; oB = xded; }

  const int wave = threadIdx.x >> 5;
  const int l = threadIdx.x & 31;
  const int h = l >> 4;
  const size_t row0 = ((size_t)blockIdx.x * 8 + wave) * 16;

  v2f a[8], bA[2][8], bB[2][8];
  load_op16x32(X + row0 * E, E, a);
  load_op16x32(Wa, E, bA[0]);
  load_op16x32(Wa + 16 * E, E, bA[1]);
  load_op16x32(Wb, E, bB[0]);
  load_op16x32(Wb + 16 * E, E, bB[1]);

#pragma unroll
  for (int nb = 0; nb < 2; ++nb) {
    const int o = (l & 15) + 16 * nb;
    v8f cA = {}; cA = wmma_chain(a, bA[nb], cA);
    v8f cB = {}; cB = wmma_chain(a, bB[nb], cB);
    const float bo = ba[o];
#pragma unroll
    for (int r = 0; r < 8; ++r) {
      const size_t m = row0 + r + 8 * h;
      oA[m * E + o] = cA[r] + bo;
      oB[m * E + o] = cB[r];
    }
  }
}

// ---------------------------------------------------------------------------
// Edge kernel: one workgroup = (b, 16 consecutive s rows, full d range).
// 8 waves x 2 s-rows each; per s-row, 32 tiles of 16 d positions.
// LDS: y_d partial accumulator (padded), den_d partial, staged xdW_ed slice.
// ---------------------------------------------------------------------------
__global__ void edge_kernel(const float* __restrict__ xe,
                            const float* __restrict__ asew, const float* __restrict__ aseb,
                            const float* __restrict__ adew, const float* __restrict__ adeb,
                            const float* __restrict__ Wee, const float* __restrict__ bee,
                            const float* __restrict__ xses, const float* __restrict__ xded,
                            float* __restrict__ he,
                            float* __restrict__ ysnum, float* __restrict__ dens,
                            float* __restrict__ ydnum, float* __restrict__ dend) {
  extern __shared__ float smem[];
  float* s_yd = smem;                               // D_DIM * LPAD
  float* s_dd = smem + D_DIM * LPAD;                // D_DIM
  float* s_xd = s_dd + D_DIM;                       // D_DIM * LPAD

  const int bs0 = blockIdx.x * 16;
  const int b = bs0 / S_DIM;
  const int s_base = bs0 % S_DIM;

  // zero y_d/den_d partials; stage xdW_ed[b,:,:] into LDS (16x reuse in WG)
  for (int i = threadIdx.x; i < D_DIM * LPAD + D_DIM; i += blockDim.x) smem[i] = 0.f;
  {
    const float* src = xded + (size_t)b * D_DIM * E;
    for (int i = threadIdx.x; i < D_DIM * E; i += blockDim.x)
      s_xd[(i >> 5) * LPAD + (i & 31)] = src[i];
  }
  __syncthreads();

  const int wave = threadIdx.x >> 5;
  const int l = threadIdx.x & 31;
  const int m = l & 15;
  const int h = l >> 4;

  v2f wee[2][8];
  load_op16x32(Wee, E, wee[0]);
  load_op16x32(Wee + 16 * E, E, wee[1]);

  v2f asw[8], adw[8];
  {
    const float* p1 = asew + 2 * h;
    const float* p2 = adew + 2 * h;
#pragma unroll
    for (int j = 0; j < 8; ++j) {
      asw[j] = *(const v2f*)(p1 + 4 * j);
      adw[j] = *(const v2f*)(p2 + 4 * j);
    }
  }
  const float seb = aseb[0], deb = adeb[0];
  float beo[2];
  beo[0] = bee[m];
  beo[1] = bee[m + 16];

  for (int row = 0; row < 2; ++row) {
    const int s = s_base + wave * 2 + row;
    const size_t bs = (size_t)b * S_DIM + s;

    float xsb[2];
    xsb[0] = xses[bs * E + m] + beo[0];
    xsb[1] = xses[bs * E + m + 16] + beo[1];

    v2f ys[8] = {};
    float den_acc = 0.f;

    for (int d0 = 0; d0 < D_DIM; d0 += 16) {
      const float* tbase = xe + (bs * D_DIM + d0) * E;
      if (d0 + 16 < D_DIM)
        __builtin_prefetch(tbase + 16 * E + (size_t)m * E + 2 * h, 0, 1);

      v2f a[8];
      load_op16x32(tbase, E, a);

      v8f c0 = {}, c1 = {};
      c0 = wmma_chain(a, wee[0], c0);
      c1 = wmma_chain(a, wee[1], c1);

      // attention logits (lane holds half the features of edge m; combine halves)
      float zs = 0.f, zd = 0.f;
#pragma unroll
      for (int j = 0; j < 8; ++j) {
        zs += a[j].x * asw[j].x + a[j].y * asw[j].y;
        zd += a[j].x * adw[j].x + a[j].y * adw[j].y;
      }
      zs += __shfl_xor(zs, 16, 32);
      zd += __shfl_xor(zd, 16, 32);
      const float es = __expf(zs + seb);
      const float ed = __expf(zd + deb);

      den_acc += es;
#pragma unroll
      for (int j = 0; j < 8; ++j) ys[j] += es * a[j];

      // y_d partial accumulation in LDS (padded stride -> conflict-free banks)
      float* ydp = s_yd + (d0 + m) * LPAD + 2 * h;
#pragma unroll
      for (int j = 0; j < 8; ++j) {
        atomicAdd(ydp + 4 * j,     ed * a[j].x);
        atomicAdd(ydp + 4 * j + 1, ed * a[j].y);
      }
      if (h == 0) atomicAdd(&s_dd[d0 + m], ed);

      // h_e = x_e @ Wee^T + b_ee + xs@Wes^T (per s) + xd@Wed^T (per d, from LDS)
      const size_t erow = bs * D_DIM + d0;
#pragma unroll
      for (int nb = 0; nb < 2; ++nb) {
        const int o = m + 16 * nb;
        const float addo = xsb[nb];
        const v8f c = nb ? c1 : c0;
#pragma unroll
        for (int r = 0; r < 8; ++r) {
          const int mm = r + 8 * h;
          const float xdv = s_xd[(d0 + mm) * LPAD + o];
          he[(erow + mm) * E + o] = c[r] + addo + xdv;
        }
      }
    }

    // reduce y_s over the 16 lanes of each half-group (m-axis reduction)
#pragma unroll
    for (int mask = 1; mask <= 8; mask <<= 1) {
#pragma unroll
      for (int j = 0; j < 8; ++j) {
        ys[j].x += __shfl_xor(ys[j].x, mask, 32);
        ys[j].y += __shfl_xor(ys[j].y, mask, 32);
      }
    }
#pragma unroll
    for (int mask = 1; mask <= 16; mask <<= 1)
      den_acc += __shfl_xor(den_acc, mask, 32);

    if (m == 0) {
      float* dst = ysnum + bs * E + 2 * h;
#pragma unroll
      for (int j = 0; j < 8; ++j) *(v2f*)(dst + 4 * j) = ys[j];
      if (h == 0) dens[bs] = den_acc * 0.5f;   // both halves counted each es
    }
  }

  __syncthreads();
  // flush LDS partials to global accumulators (cross-WG s-axis reduction)
  for (int i = threadIdx.x; i < D_DIM * E; i += blockDim.x)
    atomicAdd(&ydnum[(size_t)b * D_DIM * E + i], s_yd[(i >> 5) * LPAD + (i & 31)]);
  for (int i = threadIdx.x; i < D_DIM; i += blockDim.x)
    atomicAdd(&dend[(size_t)b * D_DIM + i], s_dd[i]);
}

// ---------------------------------------------------------------------------
// Finalize: out = base + (ynum/den) @ W^T   (blockIdx.y: 0 -> h_s, 1 -> h_d)
// ---------------------------------------------------------------------------
__global__ void fin_kernel(const float* __restrict__ ysnum, const float* __restrict__ dens,
                           const float* __restrict__ Wse, const float* __restrict__ hsb,
                           const float* __restrict__ ydnum, const float* __restrict__ dend,
                           const float* __restrict__ Wde, const float* __restrict__ hdb,
                           float* __restrict__ out) {
  const float *Y, *den, *W, *base;
  float* o_;
  if (blockIdx.y == 0) { Y = ysnum; den = dens; W = Wse; base = hsb; o_ = out; }
  else { Y = ydnum; den = dend; W = Wde; base = hdb; o_ = out + (size_t)B_DIM * S_DIM * E; }

  const int wave = threadIdx.x >> 5;
  const int l = threadIdx.x & 31;
  const int m = l & 15;
  const int h = l >> 4;
  const size_t row0 = ((size_t)blockIdx.x * 8 + wave) * 16;

  v2f a[8];
  load_op16x32(Y + row0 * E, E, a);
  const float inv = 1.f / den[row0 + m];
#pragma unroll
  for (int j = 0; j < 8; ++j) a[j] *= inv;

  v2f bw[2][8];
  load_op16x32(W, E, bw[0]);
  load_op16x32(W + 16 * E, E, bw[1]);

#pragma unroll
  for (int nb = 0; nb < 2; ++nb) {
    v8f c = {};
    c = wmma_chain(a, bw[nb], c);
    const int o = m + 16 * nb;
#pragma unroll
    for (int r = 0; r < 8; ++r) {
      const size_t rr = row0 + r + 8 * h;
      o_[rr * E + o] = c[r] + base[rr * E + o];
    }
  }
}

extern "C" void kernel_launch(void* const* d_in, const int* in_sizes, int n_in,
                              void* d_out, int out_size, void* d_ws, size_t ws_size,
                              hipStream_t stream) {
  const float* x_s   = (const float*)d_in[0];
  const float* x_d   = (const float*)d_in[1];
  const float* x_e   = (const float*)d_in[2];
  const float* W_ss  = (const float*)d_in[3];
  const float* b_ss  = (const float*)d_in[4];
  const float* W_se  = (const float*)d_in[5];
  const float* a_se_w= (const float*)d_in[6];
  const float* a_se_b= (const float*)d_in[7];
  const float* W_dd  = (const float*)d_in[8];
  const float* b_dd  = (const float*)d_in[9];
  const float* W_de  = (const float*)d_in[10];
  const float* a_de_w= (const float*)d_in[11];
  const float* a_de_b= (const float*)d_in[12];
  const float* W_ee  = (const float*)d_in[13];
  const float* b_ee  = (const float*)d_in[14];
  const float* W_es  = (const float*)d_in[15];
  const float* W_ed  = (const float*)d_in[16];

  float* out = (float*)d_out;
  float* ws = (float*)d_ws;
  const size_t NSE = (size_t)B_DIM * S_DIM * E;  // 131072

  float* hsb   = ws;
  float* xses  = ws + NSE;
  float* hdb   = ws + 2 * NSE;
  float* xded  = ws + 3 * NSE;
  float* ysnum = ws + 4 * NSE;
  float* dens  = ws + 5 * NSE;
  float* ydnum = dens + (size_t)B_DIM * S_DIM;
  float* dend  = ydnum + NSE;

  // zero the atomically-accumulated buffers (y_dnum + den_d are contiguous)
  hipMemsetAsync(ydnum, 0, (NSE + (size_t)B_DIM * D_DIM) * sizeof(float), stream);

  node_kernel<<<dim3(32, 2), 256, 0, stream>>>(x_s, x_d, W_ss, b_ss, W_es, W_dd, b_dd, W_ed,
                                               hsb, xses, hdb, xded);

  const size_t shmem = (size_t)(2 * D_DIM * LPAD + D_DIM) * sizeof(float);
  edge_kernel<<<256, 256, shmem, stream>>>(x_e, a_se_w, a_se_b, a_de_w, a_de_b, W_ee, b_ee,
                                           xses, xded, out + 2 * NSE,
                                           ysnum, dens, ydnum, dend);

  fin_kernel<<<dim3(32, 2), 256, 0, stream>>>(ysnum, dens, W_se, hsb,
                                              ydnum, dend, W_de, hdb, out);
}